// Graph_Diff_Reg_75574244541031
// MI455X (gfx1250) — compile-verified
//
#include <hip/hip_runtime.h>
#include <hip/hip_bf16.h>

#define DH 128

typedef __attribute__((ext_vector_type(2))) float v2f;
typedef __attribute__((ext_vector_type(8))) float v8f;

__device__ __forceinline__ void atomicAddF(float* p, float v) {
    __hip_atomic_fetch_add(p, v, __ATOMIC_RELAXED, __HIP_MEMORY_SCOPE_AGENT);
}

// ---------------------------------------------------------------------------
// Degree / normalization kernels
// ---------------------------------------------------------------------------
__global__ void k_init_dis(float* dis, int N) {
    int i = blockIdx.x * blockDim.x + threadIdx.x;
    if (i < N) dis[i] = 1.0f;                       // self-loop weight
}

__global__ void k_accum_deg(float* dis, const int* dst, const float* w, int E) {
    int e = blockIdx.x * blockDim.x + threadIdx.x;
    if (e < E) atomicAddF(&dis[dst[e]], w[e]);
}

__global__ void k_finalize_dis(float* dis, int N) {
    int i = blockIdx.x * blockDim.x + threadIdx.x;
    if (i < N) {
        float d = dis[i];
        dis[i] = (d > 0.0f) ? rsqrtf(d) : 0.0f;
    }
}

// ---------------------------------------------------------------------------
// FP32 WMMA GEMM: C[M,128] = A[M,128] @ B[128,128]  (row-major, exact fp32)
// One wave computes one 16x16 tile via 32x v_wmma_f32_16x16x4_f32.
// ---------------------------------------------------------------------------
__global__ __launch_bounds__(256)
void k_gemm_wmma_f32(const float* __restrict__ A, const float* __restrict__ B,
                     float* __restrict__ C, int M) {
    const int lane = threadIdx.x & 31;
    const int wave = threadIdx.x >> 5;
    const int tile = blockIdx.x * 8 + wave;          // 8 waves per block
    const int nColTiles = DH / 16;                   // 8
    const int totalTiles = (M / 16) * nColTiles;
    if (tile >= totalTiles) return;                  // wave-uniform guard

    const int rowTile = tile / nColTiles;
    const int colTile = tile % nColTiles;
    const int row0 = rowTile * 16;
    const int col0 = colTile * 16;
    const int half = lane >> 4;                      // 0 or 1
    const int l16  = lane & 15;

    v8f acc = {};
    const float* arow = A + (size_t)(row0 + l16) * DH;
#pragma unroll
    for (int k0 = 0; k0 < DH; k0 += 4) {
        v2f a, b;
        // A 16x4 tile: lane = row, VGPR j holds K = k0 + j + 2*half
        a.x = arow[k0 + 0 + 2 * half];
        a.y = arow[k0 + 1 + 2 * half];
        // B 4x16 tile: VGPR j, lanes0-15 row j, lanes16-31 row j+2; col = l16
        b.x = B[(size_t)(k0 + 0 + 2 * half) * DH + col0 + l16];
        b.y = B[(size_t)(k0 + 1 + 2 * half) * DH + col0 + l16];
        acc = __builtin_amdgcn_wmma_f32_16x16x4_f32(
                  false, a, false, b, (short)0, acc, false, false);
    }
    // C/D layout: VGPR r -> row r + 8*half, col = l16
#pragma unroll
    for (int r = 0; r < 8; ++r)
        C[(size_t)(row0 + r + 8 * half) * DH + col0 + l16] = acc[r];
}

// ---------------------------------------------------------------------------
// out[i,f] = b[f] + dis[i]^2 * h[i,f]     (bias + self-loop contribution)
// ---------------------------------------------------------------------------
__global__ void k_init_out(float* out, const float* __restrict__ h,
                           const float* __restrict__ dis,
                           const float* __restrict__ b, int N) {
    size_t tid = (size_t)blockIdx.x * blockDim.x + threadIdx.x;
    if (tid >= (size_t)N * DH) return;
    int i = (int)(tid >> 7), f = (int)(tid & (DH - 1));
    float ds = dis[i];
    out[tid] = b[f] + ds * ds * h[tid];
}

// ---------------------------------------------------------------------------
// Edge scatter: out[dst] += dis[src]*w*dis[dst] * h[src]   (one thread per
// (edge,feature); working set L2-resident, hw fp32 atomics)
// ---------------------------------------------------------------------------
__global__ void k_agg_edges(float* out, const float* __restrict__ h,
                            const int* __restrict__ src, const int* __restrict__ dst,
                            const float* __restrict__ w, const float* __restrict__ dis,
                            int E) {
    size_t tid = (size_t)blockIdx.x * blockDim.x + threadIdx.x;
    if (tid >= (size_t)E * DH) return;
    int e = (int)(tid >> 7), f = (int)(tid & (DH - 1));
    int s = src[e], d = dst[e];
    float coef = dis[s] * w[e] * dis[d];
    atomicAddF(&out[(size_t)d * DH + f], coef * h[(size_t)s * DH + f]);
}

// ---------------------------------------------------------------------------
// Pooling
// ---------------------------------------------------------------------------
__global__ void k_zero(float* p, int n) {
    int i = blockIdx.x * blockDim.x + threadIdx.x;
    if (i < n) p[i] = 0.0f;
}

__global__ void k_pool(const float* x11, const float* x12,
                       const float* x21, const float* x22,
                       const int* __restrict__ batch,
                       float* sums, float* cnt, int N) {
    size_t tid = (size_t)blockIdx.x * blockDim.x + threadIdx.x;
    if (tid >= (size_t)N * DH) return;
    int i = (int)(tid >> 7), f = (int)(tid & (DH - 1));
    float v = (x12[tid] - x11[tid]) * (x22[tid] - x21[tid]);
    int g = batch[i];
    atomicAddF(&sums[(size_t)g * DH + f], v);
    if (f == 0) atomicAddF(&cnt[g], 1.0f);
}

__global__ void k_finalize_g(float* g, const float* sums, const float* cnt) {
    int tid = blockIdx.x * blockDim.x + threadIdx.x;   // 64*128 threads
    if (tid >= 64 * DH) return;
    int i = tid >> 7;
    g[tid] = sums[tid] / fmaxf(cnt[i], 1.0f);
}

// ---------------------------------------------------------------------------
// Tiny MLP layer: out[i,j] = b[j] + sum_k in[i,k] * W[k,j]   (grid=64 rows)
// ---------------------------------------------------------------------------
__global__ void k_mlp(const float* __restrict__ in, const float* __restrict__ W,
                      const float* __restrict__ b, float* out, int Din, int Dout) {
    int i = blockIdx.x;
    int j = threadIdx.x;
    if (j >= Dout) return;
    float acc = b[j];
    for (int k = 0; k < Din; ++k)
        acc += in[(size_t)i * Din + k] * W[(size_t)k * Dout + j];
    out[(size_t)i * Dout + j] = acc;
}

// ---------------------------------------------------------------------------
static inline unsigned cdiv(size_t a, unsigned b) { return (unsigned)((a + b - 1) / b); }

extern "C" void kernel_launch(void* const* d_in, const int* in_sizes, int n_in,
                              void* d_out, int out_size, void* d_ws, size_t ws_size,
                              hipStream_t stream) {
    const int*   ei1  = (const int*)d_in[0];
    const float* ew1  = (const float*)d_in[1];
    const int*   ei2  = (const int*)d_in[2];
    const float* ew2  = (const float*)d_in[3];
    const float* fm0  = (const float*)d_in[4];
    const float* fm1  = (const float*)d_in[5];
    const int*   batch= (const int*)d_in[6];
    const float* W1   = (const float*)d_in[7];
    const float* b1   = (const float*)d_in[8];
    const float* W2   = (const float*)d_in[9];
    const float* b2   = (const float*)d_in[10];
    const float* M1w  = (const float*)d_in[11];
    const float* M1b  = (const float*)d_in[12];
    const float* M2w  = (const float*)d_in[13];
    const float* M2b  = (const float*)d_in[14];
    const float* M3w  = (const float*)d_in[15];
    const float* M3b  = (const float*)d_in[16];
    const float* M4w  = (const float*)d_in[17];
    const float* M4b  = (const float*)d_in[18];

    const int E = in_sizes[1];
    const int N = in_sizes[6];
    const int* src1 = ei1, * dst1 = ei1 + E;
    const int* src2 = ei2, * dst2 = ei2 + E;

    // workspace layout (floats)
    const size_t NF = (size_t)N * DH;
    float* ws   = (float*)d_ws;
    float* h1   = ws;
    float* h2   = h1  + NF;
    float* x11  = h2  + NF;
    float* x12  = x11 + NF;
    float* x21  = x12 + NF;
    float* x22  = x21 + NF;
    float* dis1 = x22 + NF;
    float* dis2 = dis1 + N;
    float* sums = dis2 + N;        // 64*128
    float* cnt  = sums + 64 * DH;  // 64
    float* gbuf = cnt  + 64;       // 64*128
    float* m1   = gbuf + 64 * DH;  // 64*128
    float* m2   = m1   + 64 * DH;  // 64*64
    float* m3   = m2   + 64 * 64;  // 64*32

    const unsigned T = 256;
    const unsigned gN   = cdiv(N, T);
    const unsigned gE   = cdiv(E, T);
    const unsigned gNF  = cdiv(NF, T);
    const unsigned gEF  = cdiv((size_t)E * DH, T);
    const unsigned gGemm = cdiv((size_t)(N / 16) * (DH / 16), 8);  // 8 waves/block

    // ---- degrees & symmetric normalization ----
    k_init_dis<<<gN, T, 0, stream>>>(dis1, N);
    k_init_dis<<<gN, T, 0, stream>>>(dis2, N);
    k_accum_deg<<<gE, T, 0, stream>>>(dis1, dst1, ew1, E);
    k_accum_deg<<<gE, T, 0, stream>>>(dis2, dst2, ew2, E);
    k_finalize_dis<<<gN, T, 0, stream>>>(dis1, N);
    k_finalize_dis<<<gN, T, 0, stream>>>(dis2, N);

    // ---- layer 1: h = fm @ W1 ; x1x = agg(h) + b1 ----
    k_gemm_wmma_f32<<<gGemm, T, 0, stream>>>(fm0, W1, h1, N);
    k_gemm_wmma_f32<<<gGemm, T, 0, stream>>>(fm1, W1, h2, N);
    k_init_out<<<gNF, T, 0, stream>>>(x11, h1, dis1, b1, N);
    k_init_out<<<gNF, T, 0, stream>>>(x12, h2, dis2, b1, N);
    k_agg_edges<<<gEF, T, 0, stream>>>(x11, h1, src1, dst1, ew1, dis1, E);
    k_agg_edges<<<gEF, T, 0, stream>>>(x12, h2, src2, dst2, ew2, dis2, E);

    // ---- layer 2: h = x1x @ W2 ; x2x = agg(h) + b2 ----
    k_gemm_wmma_f32<<<gGemm, T, 0, stream>>>(x11, W2, h1, N);
    k_gemm_wmma_f32<<<gGemm, T, 0, stream>>>(x12, W2, h2, N);
    k_init_out<<<gNF, T, 0, stream>>>(x21, h1, dis1, b2, N);
    k_init_out<<<gNF, T, 0, stream>>>(x22, h2, dis2, b2, N);
    k_agg_edges<<<gEF, T, 0, stream>>>(x21, h1, src1, dst1, ew1, dis1, E);
    k_agg_edges<<<gEF, T, 0, stream>>>(x22, h2, src2, dst2, ew2, dis2, E);

    // ---- global mean pool over (x12-x11)*(x22-x21) ----
    k_zero<<<cdiv(64 * DH + 64, T), T, 0, stream>>>(sums, 64 * DH + 64);
    k_pool<<<gNF, T, 0, stream>>>(x11, x12, x21, x22, batch, sums, cnt, N);
    k_finalize_g<<<cdiv(64 * DH, T), T, 0, stream>>>(gbuf, sums, cnt);

    // ---- MLP head ----
    k_mlp<<<64, 128, 0, stream>>>(gbuf, M1w, M1b, m1, DH, DH);
    k_mlp<<<64, 64,  0, stream>>>(m1,   M2w, M2b, m2, DH, DH / 2);
    k_mlp<<<64, 32,  0, stream>>>(m2,   M3w, M3b, m3, DH / 2, DH / 4);
    k_mlp<<<64, 1,   0, stream>>>(m3,   M4w, M4b, (float*)d_out, DH / 4, 1);
}